// TDFL_71021579206928
// MI455X (gfx1250) — compile-verified
//
#include <hip/hip_runtime.h>
#include <hip/hip_bf16.h>
#include <math.h>

// Problem constants from the reference: n=64, c=256, t=64, v=25, K=3, c4=64.
#define N_   64
#define C_   256
#define T_   64
#define V_   25
#define C4_  64
#define NC_  (N_ * C_)           // 16384 rows for the G branch
#define EPS_ 1e-5f

typedef float v2f __attribute__((ext_vector_type(2)));
typedef float v8f __attribute__((ext_vector_type(8)));

// ---------------------------------------------------------------------------
// Kernel 1: pooled[n,c,t] = mean over v of x[n,c,t,v]
// A wave covers 32 consecutive 25-float rows = 3200 contiguous bytes; the 25
// strided b32 loads all hit the same cachelines, so effective traffic is the
// full-rate 105 MB stream. Read with default RT hint so x parks in the 192 MB
// L2 for kernel 5.
// ---------------------------------------------------------------------------
__global__ __launch_bounds__(256)
void tdfl_pool(const float* __restrict__ x, float* __restrict__ pooled) {
  int i = blockIdx.x * blockDim.x + threadIdx.x;     // over n*c*t = 1,048,576
  if (i >= NC_ * T_) return;
  const float* p = x + (size_t)i * V_;
  float s = 0.f;
  #pragma unroll
  for (int v = 0; v < V_; ++v) s += p[v];
  pooled[i] = s * (1.0f / (float)V_);
}

// ---------------------------------------------------------------------------
// Kernel 2: G branch. One wave per 16-row tile of pooled (16384 x 64).
//   H = P @ Wg1^T (64->128), BN+ReLU, logits = H @ Wg2^T (128->3), softmax.
// V_WMMA_F32_16X16X4_F32 layouts per ISA 7.12.2:
//   A 16x4 f32 : lane%16 = row M; vgpr {0,1} = K{0,1} (lanes<16) / K{2,3}
//   B  4x16 f32: lane%16 = col N; vgpr {0,1} = K rows, same half split
//   C/D        : vgpr r -> M = r + 8*(lane>=16), lane%16 = N
// A tile is preloaded once (32 VGPRs); per-lane logit partials accumulate
// across all 8 N-tiles and are butterfly-reduced exactly once at the end.
// ---------------------------------------------------------------------------
__global__ __launch_bounds__(32)
void tdfl_gbranch(const float* __restrict__ pooled,   // (16384, 64)
                  const float* __restrict__ Wg1,      // (128, 64) row-major
                  const float* __restrict__ gamma1, const float* __restrict__ beta1,
                  const float* __restrict__ mean1,  const float* __restrict__ var1,
                  const float* __restrict__ Wg2,      // (3, 128) row-major
                  float* __restrict__ kern)           // (16384, 3)
{
  const int lane = threadIdx.x;
  const int half = lane >> 4;          // 0 or 1
  const int l16  = lane & 15;
  const int m0   = blockIdx.x * 16;    // row-tile base, 1024 blocks

  // Preload the whole A panel (this wave's 16 rows x 64 K) once.
  v2f a[16];
  {
    const float* arow = pooled + (size_t)(m0 + l16) * T_ + half * 2;
    #pragma unroll
    for (int kk = 0; kk < 16; ++kk) {  // pairs merge into global_load_b64
      a[kk].x = arow[kk * 4];
      a[kk].y = arow[kk * 4 + 1];
    }
  }

  float lg[8][3];
  #pragma unroll
  for (int r = 0; r < 8; ++r)
    for (int q = 0; q < 3; ++q) lg[r][q] = 0.f;

  #pragma unroll 1
  for (int j = 0; j < 8; ++j) {        // 8 N-tiles covering 128 hidden cols
    v8f c = {};
    const float* brow = Wg1 + (size_t)(j * 16 + l16) * T_ + half * 2;
    #pragma unroll
    for (int kk = 0; kk < 16; ++kk) {  // K = 64 in steps of 4
      v2f b;
      b.x = brow[kk * 4];
      b.y = brow[kk * 4 + 1];
      c = __builtin_amdgcn_wmma_f32_16x16x4_f32(false, a[kk], false, b,
                                                (short)0, c, false, false);
    }
    // BN + ReLU per hidden column, then per-lane partial logit products.
    const int col = j * 16 + l16;
    const float sc = gamma1[col] * rsqrtf(var1[col] + EPS_);
    const float sh = beta1[col] - mean1[col] * sc;
    const float w0 = Wg2[0 * 128 + col];
    const float w1 = Wg2[1 * 128 + col];
    const float w2 = Wg2[2 * 128 + col];
    #pragma unroll
    for (int r = 0; r < 8; ++r) {
      float h = fmaxf(c[r] * sc + sh, 0.f);
      lg[r][0] += h * w0;
      lg[r][1] += h * w1;
      lg[r][2] += h * w2;
    }
  }

  // Single butterfly reduction over the 16 columns held by each half-wave
  // (masks < 16 keep the exchange inside each 16-lane group, preserving the
  // distinct M rows of the two halves).
  #pragma unroll
  for (int r = 0; r < 8; ++r)
    for (int q = 0; q < 3; ++q) {
      float p = lg[r][q];
      #pragma unroll
      for (int m = 1; m < 16; m <<= 1) p += __shfl_xor(p, m, 32);
      lg[r][q] = p;
    }

  if (l16 == 0) {                      // lane 0 / lane 16 own their 8 rows
    #pragma unroll
    for (int r = 0; r < 8; ++r) {
      const int row = m0 + r + half * 8;
      float a0 = lg[r][0], a1 = lg[r][1], a2 = lg[r][2];
      float mx = fmaxf(a0, fmaxf(a1, a2));
      float e0 = __expf(a0 - mx), e1 = __expf(a1 - mx), e2 = __expf(a2 - mx);
      float inv = 1.f / (e0 + e1 + e2);
      kern[row * 3 + 0] = e0 * inv;
      kern[row * 3 + 1] = e1 * inv;
      kern[row * 3 + 2] = e2 * inv;
    }
  }
}

// ---------------------------------------------------------------------------
// Kernel 3: L branch conv1 (c -> c/4, k=3, pad=1) + BN + ReLU, as WMMA GEMM.
// Block per n (64 blocks, 512 threads = 16 waves). pooled[n] staged into LDS
// in a PAIR-INTERLEAVED zero-padded layout lds[pair=ci/2][tcol][ci&1] so each
// lane's (K, K+1) B operand is one contiguous 8-byte LDS read (ds_load_b64).
//   a[n, co, t] = relu(bn( sum_k sum_ci WL1[co,ci,k] * pooledPad[n,ci,t+k] ))
// ---------------------------------------------------------------------------
__global__ __launch_bounds__(512)
void tdfl_lconv1(const float* __restrict__ pooled,   // (n, 256, 64)
                 const float* __restrict__ WL1,      // (64, 256, 3) row-major
                 const float* __restrict__ g2, const float* __restrict__ b2,
                 const float* __restrict__ m2, const float* __restrict__ v2,
                 float* __restrict__ a_out)          // (n, 64, 64)
{
  __shared__ float ldsB[(C_ / 2) * 66 * 2];          // 67.6 KB of 320 KB WGP LDS
  const int n   = blockIdx.x;
  const int tid = threadIdx.x;

  const float* pn = pooled + (size_t)n * C_ * T_;
  for (int i = tid; i < C_ * T_; i += 512) {
    const int ci = i >> 6, t = i & 63;
    ldsB[((ci >> 1) * 66 + t + 1) * 2 + (ci & 1)] = pn[i];
  }
  for (int p = tid; p < C_ / 2; p += 512) {          // temporal zero padding
    ldsB[(p * 66 + 0)  * 2 + 0] = 0.f;
    ldsB[(p * 66 + 0)  * 2 + 1] = 0.f;
    ldsB[(p * 66 + 65) * 2 + 0] = 0.f;
    ldsB[(p * 66 + 65) * 2 + 1] = 0.f;
  }
  __syncthreads();

  const int wave = tid >> 5;                         // 0..15
  const int lane = tid & 31;
  const int half = lane >> 4, l16 = lane & 15;
  const int mi = wave >> 2, nj = wave & 3;
  const v2f* ldsB2 = (const v2f*)ldsB;

  v8f c = {};
  #pragma unroll 1
  for (int k = 0; k < 3; ++k) {                      // 3 temporal taps
    const float* wrow = WL1 + ((size_t)(mi * 16 + l16) * C_) * 3 + k;
    const int tcol = nj * 16 + l16 + k;              // padded B column
    #pragma unroll 2
    for (int kk = 0; kk < 64; ++kk) {                // K = 256 in steps of 4
      const int cb = kk * 4 + half * 2;              // first K row of this lane
      v2f a;
      a.x = wrow[(size_t)cb * 3];
      a.y = wrow[(size_t)(cb + 1) * 3];
      const v2f b = ldsB2[(kk * 2 + half) * 66 + tcol];  // rows cb, cb+1
      c = __builtin_amdgcn_wmma_f32_16x16x4_f32(false, a, false, b,
                                                (short)0, c, false, false);
    }
  }
  // BN (per output channel = row M) + ReLU, write a tile.
  #pragma unroll
  for (int r = 0; r < 8; ++r) {
    const int co = mi * 16 + r + half * 8;
    const float sc = g2[co] * rsqrtf(v2[co] + EPS_);
    const float sh = b2[co] - m2[co] * sc;
    const float h  = fmaxf(c[r] * sc + sh, 0.f);
    a_out[((size_t)n * C4_ + co) * T_ + nj * 16 + l16] = h;
  }
}

// ---------------------------------------------------------------------------
// Kernel 4: 1x1 conv (c/4 -> c) + sigmoid.  act[n,co,t] = sigma(sum_ci W*a)
// ---------------------------------------------------------------------------
__global__ __launch_bounds__(256)
void tdfl_lconv2(const float* __restrict__ a_in,     // (n, 64, 64)
                 const float* __restrict__ WL2,      // (256, 64)
                 float* __restrict__ act)            // (n, 256, 64)
{
  int idx = blockIdx.x * blockDim.x + threadIdx.x;   // n*256*64 = 1,048,576
  if (idx >= N_ * C_ * T_) return;
  const int t  = idx & 63;
  const int co = (idx >> 6) & 255;
  const int n  = idx >> 14;
  const float* ap = a_in + (size_t)n * C4_ * T_ + t;
  const float* wp = WL2 + co * C4_;
  float s = 0.f;
  #pragma unroll
  for (int ci = 0; ci < C4_; ++ci) s += wp[ci] * ap[ci * T_];
  act[idx] = 1.f / (1.f + __expf(-s));
}

// ---------------------------------------------------------------------------
// Kernel 5: fused gating + dynamic depthwise temporal conv + residual.
//   out[n,c,t,v] = x[n,c,t,v] + sum_k kern[n,c,k]*act[n,c,t+k-1]*x[n,c,t+k-1,v]
// Block per (n,c): x tile in LDS so x is read exactly once here (and it should
// still be L2-resident from kernel 1: 105 MB vs 192 MB L2). out is written
// with NON-TEMPORAL stores so the 105 MB output stream does not evict x.
// ---------------------------------------------------------------------------
__global__ __launch_bounds__(256)
void tdfl_final(const float* __restrict__ x,
                const float* __restrict__ act,       // (n, 256, 64)
                const float* __restrict__ kern,      // (n*256, 3)
                float* __restrict__ out)
{
  __shared__ float lx[T_ * V_];                      // 6.4 KB
  __shared__ float wk[3][T_];                        // kern[k]*act[t]
  const int nc  = blockIdx.x;                        // n*256 + c
  const int tid = threadIdx.x;

  const float* xp = x + (size_t)nc * T_ * V_;
  __builtin_prefetch(xp + tid * 8, 0, 0);            // global_prefetch_b8 warmup
  for (int i = tid; i < T_ * V_; i += 256) lx[i] = xp[i];
  if (tid < T_) {
    const float a  = act[(size_t)nc * T_ + tid];
    const float k0 = kern[nc * 3 + 0];
    const float k1 = kern[nc * 3 + 1];
    const float k2 = kern[nc * 3 + 2];
    wk[0][tid] = k0 * a; wk[1][tid] = k1 * a; wk[2][tid] = k2 * a;
  }
  __syncthreads();

  float* op = out + (size_t)nc * T_ * V_;
  for (int i = tid; i < T_ * V_; i += 256) {
    const int t = i / V_;
    const int v = i - t * V_;
    float acc = lx[i];                               // residual
    #pragma unroll
    for (int k = 0; k < 3; ++k) {
      const int tt = t + k - 1;
      if (tt >= 0 && tt < T_)
        acc += wk[k][tt] * lx[tt * V_ + v];
    }
    __builtin_nontemporal_store(acc, op + i);        // store TH=NT
  }
}

// ---------------------------------------------------------------------------
extern "C" void kernel_launch(void* const* d_in, const int* in_sizes, int n_in,
                              void* d_out, int out_size, void* d_ws, size_t ws_size,
                              hipStream_t stream) {
  const float* x      = (const float*)d_in[0];
  const float* Wg1    = (const float*)d_in[1];
  const float* gamma1 = (const float*)d_in[2];
  const float* beta1  = (const float*)d_in[3];
  const float* mean1  = (const float*)d_in[4];
  const float* var1   = (const float*)d_in[5];
  const float* Wg2    = (const float*)d_in[6];
  const float* WL1    = (const float*)d_in[7];
  const float* gamma2 = (const float*)d_in[8];
  const float* beta2  = (const float*)d_in[9];
  const float* mean2  = (const float*)d_in[10];
  const float* var2   = (const float*)d_in[11];
  const float* WL2    = (const float*)d_in[12];
  float* out = (float*)d_out;

  // Workspace layout (all intermediates tiny -> L2-resident):
  char* ws = (char*)d_ws;
  float* pooled = (float*)(ws);                                  // 4 MB
  float* kern   = (float*)(ws + (size_t)NC_ * T_ * 4);           // 192 KB
  float* a_ws   = (float*)(ws + (size_t)NC_ * T_ * 4
                              + (size_t)NC_ * 3 * 4);            // 1 MB
  float* act    = (float*)(ws + (size_t)NC_ * T_ * 4
                              + (size_t)NC_ * 3 * 4
                              + (size_t)N_ * C4_ * T_ * 4);      // 4 MB

  // 1) pool over joints (streams x into L2)
  tdfl_pool<<<(NC_ * T_ + 255) / 256, 256, 0, stream>>>(x, pooled);
  // 2) G branch: per-(n,c) temporal kernel (WMMA f32)
  tdfl_gbranch<<<NC_ / 16, 32, 0, stream>>>(pooled, Wg1, gamma1, beta1,
                                            mean1, var1, Wg2, kern);
  // 3) L branch conv1 + BN + ReLU (WMMA f32, LDS-staged)
  tdfl_lconv1<<<N_, 512, 0, stream>>>(pooled, WL1, gamma2, beta2,
                                      mean2, var2, a_ws);
  // 4) L branch 1x1 conv + sigmoid
  tdfl_lconv2<<<(N_ * C_ * T_ + 255) / 256, 256, 0, stream>>>(a_ws, WL2, act);
  // 5) gating + dynamic depthwise temporal conv + residual (NT stores)
  tdfl_final<<<NC_, 256, 0, stream>>>(x, act, kern, out);
}